// RecommendationSystem_36953898615067
// MI455X (gfx1250) — compile-verified
//
#include <hip/hip_runtime.h>
#include <hip/hip_bf16.h>
#include <math.h>

typedef __attribute__((ext_vector_type(2))) float v2f;
typedef __attribute__((ext_vector_type(8))) float v8f;

#define DIM   64
#define DIM2  128
#define MSG_EPS 1e-7f
#define BN_EPS  1e-5f

// ---------------------------------------------------------------------------
// Edge phase 1: per-destination max of m = relu(x_src[src]) + eps.
// m > 0 always, so u32 bit-pattern atomicMax == float max.
__global__ void genconv_msgmax(const float* __restrict__ xsrc,
                               const int* __restrict__ sidx,
                               const int* __restrict__ didx,
                               unsigned int* __restrict__ mx, int E) {
    int idx = blockIdx.x * blockDim.x + threadIdx.x;     // E*64 <= 64M < 2^31
    if (idx >= E * DIM) return;
    int e = idx >> 6, f = idx & 63;
    int s = sidx[e], d = didx[e];
    float m = fmaxf(xsrc[(long long)s * DIM + f], 0.0f) + MSG_EPS;
    atomicMax(&mx[(long long)d * DIM + f], __float_as_uint(m));
}

// Edge phase 2: den = sum exp(m - mx), num = sum exp(m - mx) * m per dst.
__global__ void genconv_expsum(const float* __restrict__ xsrc,
                               const int* __restrict__ sidx,
                               const int* __restrict__ didx,
                               const float* __restrict__ mx,
                               float* __restrict__ den,
                               float* __restrict__ num, int E) {
    int idx = blockIdx.x * blockDim.x + threadIdx.x;
    if (idx >= E * DIM) return;
    int e = idx >> 6, f = idx & 63;
    int s = sidx[e], d = didx[e];
    long long di = (long long)d * DIM + f;
    float m  = fmaxf(xsrc[(long long)s * DIM + f], 0.0f) + MSG_EPS;
    float ex = expf(m - mx[di]);
    atomicAdd(&den[di], ex);
    atomicAdd(&num[di], ex * m);
}

// h0 = softmax-weighted aggregate + root skip
__global__ void genconv_agg(const float* __restrict__ num,
                            const float* __restrict__ den,
                            const float* __restrict__ xdst,
                            float* __restrict__ h0, long long total) {
    long long i = (long long)blockIdx.x * blockDim.x + threadIdx.x;
    if (i >= total) return;
    float dn = den[i];
    float a  = (dn > 0.0f) ? (num[i] / dn) : 0.0f;
    h0[i] = a + xdst[i];
}

// ---------------------------------------------------------------------------
// GEMM1: H1[N,128] = h0[N,64] @ W1[64,128] + b1.  256 threads = 8 waves;
// each wave owns one 16-col tile of one 16-row tile; K-loop of 16 WMMAs.
__global__ void genconv_gemm1(const float* __restrict__ X,   // [N,64]
                              const float* __restrict__ W,   // [64,128]
                              const float* __restrict__ bias,// [128]
                              float* __restrict__ H, int N) {
    int rowTile = blockIdx.x * 16;
    int wave = threadIdx.x >> 5;          // 0..7 -> col tile
    int lane = threadIdx.x & 31;
    int colTile = wave * 16;
    int mrow  = lane & 15;
    int khalf = (lane >> 4) * 2;          // 0 or 2 (A/B 16x4 f32 layout)
    int arow  = min(rowTile + mrow, N - 1);
    v8f c = {};
    for (int k = 0; k < DIM; k += 4) {
        int kk = k + khalf;
        v2f a, b;
        const float* xr = X + (long long)arow * DIM + kk;
        a.x = xr[0]; a.y = xr[1];
        const float* wr = W + (long long)kk * DIM2 + colTile + mrow;
        b.x = wr[0]; b.y = wr[DIM2];
        c = __builtin_amdgcn_wmma_f32_16x16x4_f32(false, a, false, b,
                                                  (short)0, c, false, false);
    }
    int col = colTile + (lane & 15);
    float bb = bias[col];
    for (int v = 0; v < 8; ++v) {
        int row = rowTile + v + (lane >> 4) * 8;
        if (row < N) H[(long long)row * DIM2 + col] = c[v] + bb;
    }
}

// BatchNorm column statistics. Launch with (grid*block) % 128 == 0 so every
// thread stays on a single column while striding.
__global__ void genconv_bnstats(const float* __restrict__ H,
                                float* __restrict__ sums,
                                float* __restrict__ sumsq, long long total) {
    long long stride = (long long)gridDim.x * blockDim.x;
    long long i = (long long)blockIdx.x * blockDim.x + threadIdx.x;
    int col = (int)(i & (DIM2 - 1));
    float s = 0.0f, q = 0.0f;
    for (; i < total; i += stride) {
        float v = H[i];
        s += v; q += v * v;
    }
    atomicAdd(&sums[col], s);
    atomicAdd(&sumsq[col], q);
}

__global__ void genconv_bnfinal(const float* __restrict__ sums,
                                const float* __restrict__ sumsq,
                                const float* __restrict__ gamma,
                                const float* __restrict__ beta,
                                float* __restrict__ scale,
                                float* __restrict__ shift, int N) {
    int c = threadIdx.x;                  // 128 threads
    float invN = 1.0f / (float)N;
    float mu  = sums[c] * invN;
    float var = sumsq[c] * invN - mu * mu;
    float sc  = gamma[c] * rsqrtf(var + BN_EPS);
    scale[c] = sc;
    shift[c] = beta[c] - mu * sc;
}

// GEMM2: Y[N,64] = relu( relu(BN(H1)) @ W2[128,64] + b2 ).  BN+ReLU fused
// into the A-fragment load; outer ReLU fused into the store.
__global__ void genconv_gemm2(const float* __restrict__ H,   // [N,128]
                              const float* __restrict__ W,   // [128,64]
                              const float* __restrict__ bias,// [64]
                              const float* __restrict__ scale,
                              const float* __restrict__ shift,
                              float* __restrict__ Y, int N) {
    int rowTile = blockIdx.x * 16;
    int wave = threadIdx.x >> 5;          // 0..3 -> col tile (64 cols)
    int lane = threadIdx.x & 31;
    int colTile = wave * 16;
    int mrow  = lane & 15;
    int khalf = (lane >> 4) * 2;
    int arow  = min(rowTile + mrow, N - 1);
    v8f c = {};
    for (int k = 0; k < DIM2; k += 4) {
        int kk = k + khalf;
        v2f a, b;
        const float* hr = H + (long long)arow * DIM2 + kk;
        a.x = fmaxf(hr[0] * scale[kk]     + shift[kk],     0.0f);
        a.y = fmaxf(hr[1] * scale[kk + 1] + shift[kk + 1], 0.0f);
        const float* wr = W + (long long)kk * DIM + colTile + mrow;
        b.x = wr[0]; b.y = wr[DIM];
        c = __builtin_amdgcn_wmma_f32_16x16x4_f32(false, a, false, b,
                                                  (short)0, c, false, false);
    }
    int col = colTile + (lane & 15);
    float bb = bias[col];
    for (int v = 0; v < 8; ++v) {
        int row = rowTile + v + (lane >> 4) * 8;
        if (row < N) Y[(long long)row * DIM + col] = fmaxf(c[v] + bb, 0.0f);
    }
}

// Decoder: one wave32 per supervision edge, 2 features per lane.
__global__ void decode_dot(const float* __restrict__ xu,
                           const float* __restrict__ xi,
                           const int* __restrict__ pu,
                           const int* __restrict__ pi,
                           float* __restrict__ out, int B) {
    int gid  = blockIdx.x * blockDim.x + threadIdx.x;
    int edge = gid >> 5;
    int lane = threadIdx.x & 31;
    if (edge >= B) return;
    const float* u = xu + (long long)pu[edge] * DIM;
    const float* v = xi + (long long)pi[edge] * DIM;
    float p = u[lane] * v[lane] + u[lane + 32] * v[lane + 32];
    for (int o = 16; o > 0; o >>= 1) p += __shfl_xor(p, o, 32);
    if (lane == 0) out[edge] = p;
}

// ---------------------------------------------------------------------------
extern "C" void kernel_launch(void* const* d_in, const int* in_sizes, int n_in,
                              void* d_out, int out_size, void* d_ws, size_t ws_size,
                              hipStream_t stream) {
    const float* x_user = (const float*)d_in[0];
    const float* x_item = (const float*)d_in[1];
    const float* W1     = (const float*)d_in[2];   // (3,2,64,128)
    const float* b1     = (const float*)d_in[3];   // (3,2,128)
    const float* gamma  = (const float*)d_in[4];   // (3,2,128)
    const float* beta   = (const float*)d_in[5];   // (3,2,128)
    const float* W2     = (const float*)d_in[6];   // (3,2,128,64)
    const float* b2     = (const float*)d_in[7];   // (3,2,64)
    const int* edge_ui  = (const int*)d_in[8];     // (2,E)
    const int* pred     = (const int*)d_in[9];     // (2,B)

    const int NU = in_sizes[0] / DIM;
    const int NI = in_sizes[1] / DIM;
    const int E  = in_sizes[8] / 2;
    const int B  = in_sizes[9] / 2;
    const int* src_u = edge_ui;
    const int* dst_i = edge_ui + E;
    const int* pu = pred;
    const int* pi = pred + B;

    // Workspace layout (floats)
    float* ws = (float*)d_ws;
    size_t NU_D = (size_t)NU * DIM, NI_D = (size_t)NI * DIM;
    float* xu0 = ws; ws += NU_D;
    float* xi0 = ws; ws += NI_D;
    float* xu1 = ws; ws += NU_D;
    float* xi1 = ws; ws += NI_D;
    float* mx  = ws; ws += NU_D;               // sized for max dst count
    float* den = ws; ws += NU_D;
    float* num = ws; ws += NU_D;
    float* h0  = ws; ws += NU_D;
    float* H1  = ws; ws += (size_t)NU * DIM2;
    float* sums  = ws; ws += DIM2;
    float* sumsq = ws; ws += DIM2;
    float* scale = ws; ws += DIM2;
    float* shift = ws; ws += DIM2;

    auto conv = [&](const float* xsrc, const float* xdst,
                    const int* sidx, const int* didx, int Ndst,
                    const float* W1l, const float* b1l,
                    const float* gl, const float* btl,
                    const float* W2l, const float* b2l, float* out) {
        size_t nd = (size_t)Ndst * DIM;
        hipMemsetAsync(mx,  0, nd * sizeof(float), stream);
        hipMemsetAsync(den, 0, nd * sizeof(float), stream);
        hipMemsetAsync(num, 0, nd * sizeof(float), stream);
        int eblocks = (E * DIM + 255) / 256;
        genconv_msgmax<<<eblocks, 256, 0, stream>>>(xsrc, sidx, didx,
                                                    (unsigned int*)mx, E);
        genconv_expsum<<<eblocks, 256, 0, stream>>>(xsrc, sidx, didx,
                                                    mx, den, num, E);
        genconv_agg<<<(int)((nd + 255) / 256), 256, 0, stream>>>(
            num, den, xdst, h0, (long long)nd);
        int rtiles = (Ndst + 15) / 16;
        genconv_gemm1<<<rtiles, 256, 0, stream>>>(h0, W1l, b1l, H1, Ndst);
        hipMemsetAsync(sums,  0, DIM2 * sizeof(float), stream);
        hipMemsetAsync(sumsq, 0, DIM2 * sizeof(float), stream);
        genconv_bnstats<<<512, 256, 0, stream>>>(H1, sums, sumsq,
                                                 (long long)Ndst * DIM2);
        genconv_bnfinal<<<1, DIM2, 0, stream>>>(sums, sumsq, gl, btl,
                                                scale, shift, Ndst);
        genconv_gemm2<<<rtiles, 128, 0, stream>>>(H1, W2l, b2l, scale, shift,
                                                  out, Ndst);
    };

    const float* cu = x_user;
    const float* ci = x_item;
    float* ou = xu0;
    float* oi = xi0;
    for (int l = 0; l < 3; ++l) {
        size_t t0 = (size_t)(l * 2 + 0);
        size_t t1 = (size_t)(l * 2 + 1);
        // edge type 0: user -> item (dst = items)
        conv(cu, ci, src_u, dst_i, NI,
             W1 + t0 * DIM * DIM2, b1 + t0 * DIM2,
             gamma + t0 * DIM2, beta + t0 * DIM2,
             W2 + t0 * DIM2 * DIM, b2 + t0 * DIM, oi);
        // edge type 1: item -> user (dst = users)
        conv(ci, cu, dst_i, src_u, NU,
             W1 + t1 * DIM * DIM2, b1 + t1 * DIM2,
             gamma + t1 * DIM2, beta + t1 * DIM2,
             W2 + t1 * DIM2 * DIM, b2 + t1 * DIM, ou);
        cu = ou; ci = oi;
        if (l == 0)      { ou = xu1; oi = xi1; }
        else             { ou = (ou == xu1) ? xu0 : xu1;
                           oi = (oi == xi1) ? xi0 : xi1; }
    }

    int dblocks = (B * 32 + 255) / 256;
    decode_dot<<<dblocks, 256, 0, stream>>>(cu, ci, pu, pi, (float*)d_out, B);
}